// S_ATT_85332410237225
// MI455X (gfx1250) — compile-verified
//
#include <hip/hip_runtime.h>
#include <hip/hip_bf16.h>
#include <cstdint>

typedef float v2f __attribute__((ext_vector_type(2)));
typedef float v8f __attribute__((ext_vector_type(8)));
typedef int   v4i __attribute__((ext_vector_type(4)));

#define BB   16
#define LL   512
#define DD   512
#define HEADS 8
#define HD   64

// ---------------------------------------------------------------------------
// Async global->LDS staging (CDNA5 GLOBAL_LOAD_ASYNC_TO_LDS_B128, ASYNCcnt).
// Builtin signature (from clang diagnostic): (v4i AS1*, v4i AS3*, imm, imm).
// AS casts done via inttoptr: AS1 is identity-mapped for global generic
// pointers; AS3 is the low 32 bits of a generic LDS pointer.
// ---------------------------------------------------------------------------
#if defined(__has_builtin)
#if __has_builtin(__builtin_amdgcn_global_load_async_to_lds_b128) && \
    __has_builtin(__builtin_amdgcn_s_wait_asynccnt)
#define USE_ASYNC_LDS 1
#endif
#endif

__device__ __forceinline__ void stage16(const float* g, float* l) {
#ifdef USE_ASYNC_LDS
  __builtin_amdgcn_global_load_async_to_lds_b128(
      (__attribute__((address_space(1))) v4i*)(uintptr_t)g,
      (__attribute__((address_space(3))) v4i*)(uint32_t)(uintptr_t)l, 0, 0);
#else
  *(float4*)l = *(const float4*)g;
#endif
}

__device__ __forceinline__ void stage_wait() {
#ifdef USE_ASYNC_LDS
  __builtin_amdgcn_s_wait_asynccnt(0);
#endif
}

// ---------------------------------------------------------------------------
// GEMM: C[m,n] = act( sum_k A[m,k]*W[n,k] + bias[n] (+ resid[m,n]) )
// Block tile 128(M) x 16(N), K chunked by 64 through LDS, fp32 WMMA 16x16x4.
// Frag layout (CDNA5 ISA 7.12.2); both frags gathered as tile[i][k] row-major
// computes A * W^T exactly in fp32 (matches the fp32 reference).
// ---------------------------------------------------------------------------
template <bool RELU, bool RESID>
__global__ __launch_bounds__(256) void gemm_wmma_kernel(
    const float* __restrict__ A, const float* __restrict__ W,
    const float* __restrict__ bias, const float* __restrict__ resid,
    float* __restrict__ C, int M, int N, int K) {
  __shared__ __align__(16) float As[128][68];
  __shared__ __align__(16) float Bs[16][68];

  const int t    = threadIdx.x;
  const int wave = t >> 5;
  const int lane = t & 31;
  const int lm   = lane & 15;
  const int lk   = (lane >> 4) << 1;  // 0 or 2
  const int m0   = blockIdx.x * 128;
  const int n0   = blockIdx.y * 16;

  v8f acc = {0.f, 0.f, 0.f, 0.f, 0.f, 0.f, 0.f, 0.f};

  for (int kc = 0; kc < K; kc += 64) {
    // stage A: 128x64 floats = 2048 x b128, 8 per thread (coalesced)
#pragma unroll
    for (int i = 0; i < 8; ++i) {
      int idx = t + i * 256;
      int r = idx >> 4, c4 = (idx & 15) << 2;
      stage16(A + (size_t)(m0 + r) * K + kc + c4, &As[r][c4]);
    }
    // stage W tile: 16x64 = 256 x b128, 1 per thread
    {
      int r = t >> 4, c4 = (t & 15) << 2;
      stage16(W + (size_t)(n0 + r) * K + kc + c4, &Bs[r][c4]);
    }
    if (kc + 64 < K)  // hint next chunk into cache (global_prefetch_b8)
      __builtin_prefetch(A + (size_t)(m0 + (t >> 4)) * K + kc + 64, 0, 1);
    stage_wait();
    __syncthreads();

#pragma unroll
    for (int kk = 0; kk < 64; kk += 4) {
      v2f a, b;
      a.x = As[(wave << 4) + lm][kk + lk];
      a.y = As[(wave << 4) + lm][kk + lk + 1];
      b.x = Bs[lm][kk + lk];
      b.y = Bs[lm][kk + lk + 1];
      acc = __builtin_amdgcn_wmma_f32_16x16x4_f32(false, a, false, b,
                                                  (short)0, acc, false, false);
    }
    __syncthreads();
  }

  const float bcol = bias[n0 + lm];
#pragma unroll
  for (int r = 0; r < 8; ++r) {
    int row = m0 + (wave << 4) + r + ((lane >> 4) << 3);
    int col = n0 + lm;
    float v = acc[r] + bcol;
    if (RESID) v += resid[(size_t)row * N + col];
    if (RELU) v = fmaxf(v, 0.f);
    C[(size_t)row * N + col] = v;
  }
}

// ---------------------------------------------------------------------------
// alpha[b,h] = max(sigmoid(mq[b,h,L-1,:] . Wa + ba) + 1, 1 + 1e-5)
// ---------------------------------------------------------------------------
__global__ void alpha_kernel(const float* __restrict__ q,
                             const float* __restrict__ Wa,
                             const float* __restrict__ ba,
                             float* __restrict__ alphaBH) {
  int t = threadIdx.x;  // 0..127 : (b,h)
  int b = t >> 3, h = t & 7;
  const float* row = q + ((size_t)(b * LL + (LL - 1)) * DD) + h * HD;
  float z = ba[0];
#pragma unroll 8
  for (int j = 0; j < HD; ++j) z += row[j] * Wa[j];
  float a = 1.f / (1.f + expf(-z)) + 1.f;
  alphaBH[t] = fmaxf(a, 1.f + 1e-5f);
}

// ---------------------------------------------------------------------------
// Attention + entmax-bisect per (b, h, 16-query tile). 128 threads = 4 waves.
//   phase 1: scores tile [16 x 512] via WMMA, scaled + branchless LDS mask
//   phase 2: 50-iter entmax bisection per row, in registers (16 elems/lane)
//   phase 3: att tile [16 x 64] = p @ mk via WMMA
// ---------------------------------------------------------------------------
__global__ __launch_bounds__(128) void attn_entmax_kernel(
    const float* __restrict__ q, const float* __restrict__ X,
    const float* __restrict__ mask, const float* __restrict__ alphaBH,
    float* __restrict__ att) {
  __shared__ __align__(16) float Sq[16][68];   // mq tile
  __shared__ __align__(16) float Sk[64][68];   // staged keys (mk = X slice)
  __shared__ float Sc[16][516];                // scores -> p
  __shared__ float Sm[512];                    // mask row (key 511 forced 0)

  const int b  = blockIdx.z;
  const int h  = blockIdx.y;
  const int q0 = blockIdx.x * 16;
  const int t = threadIdx.x, wave = t >> 5, lane = t & 31;
  const int lm = lane & 15, lk = (lane >> 4) << 1;

  // stage mask row once (branchless epilogue later)
#pragma unroll
  for (int i = 0; i < 4; ++i) {
    int k = t + i * 128;
    Sm[k] = (k < LL - 1) ? mask[b * (LL - 1) + k] : 0.f;
  }
  // stage mq tile: 16x64 = 256 x b128
#pragma unroll
  for (int i = 0; i < 2; ++i) {
    int idx = t + i * 128;
    int r = idx >> 4, c4 = (idx & 15) << 2;
    stage16(q + (size_t)(b * LL + q0 + r) * DD + h * HD + c4, &Sq[r][c4]);
  }

  // ---- phase 1: scores = mq . mk^T / 8, masked ----
  for (int kb = 0; kb < LL; kb += 64) {
    __syncthreads();
#pragma unroll
    for (int i = 0; i < 8; ++i) {
      int idx = t + i * 128;
      int r = idx >> 4, c4 = (idx & 15) << 2;
      stage16(X + (size_t)(b * LL + kb + r) * DD + h * HD + c4, &Sk[r][c4]);
    }
    stage_wait();
    __syncthreads();

    v8f acc = {0.f, 0.f, 0.f, 0.f, 0.f, 0.f, 0.f, 0.f};
#pragma unroll
    for (int kk = 0; kk < HD; kk += 4) {
      v2f a, bb;
      a.x  = Sq[lm][kk + lk];
      a.y  = Sq[lm][kk + lk + 1];
      bb.x = Sk[(wave << 4) + lm][kk + lk];
      bb.y = Sk[(wave << 4) + lm][kk + lk + 1];
      acc = __builtin_amdgcn_wmma_f32_16x16x4_f32(false, a, false, bb,
                                                  (short)0, acc, false, false);
    }
#pragma unroll
    for (int r = 0; r < 8; ++r) {
      int qr  = r + ((lane >> 4) << 3);
      int key = kb + (wave << 4) + lm;
      float s = acc[r] * 0.125f;  // 1/sqrt(64)
      Sc[qr][key] = (Sm[key] != 0.f) ? s : -__builtin_huge_valf();
    }
  }
  __syncthreads();

  // ---- phase 2: entmax bisection (one wave handles 4 rows) ----
  const float am1  = alphaBH[b * HEADS + h] - 1.f;
  const float rinv = 1.f / am1;
  for (int rr = 0; rr < 4; ++rr) {
    const int row = wave * 4 + rr;
    float xs[16];
#pragma unroll
    for (int j = 0; j < 16; ++j) xs[j] = Sc[row][lane + 32 * j] * am1;

    float mx = -__builtin_huge_valf();
#pragma unroll
    for (int j = 0; j < 16; ++j) mx = fmaxf(mx, xs[j]);
#pragma unroll
    for (int off = 16; off; off >>= 1) mx = fmaxf(mx, __shfl_xor(mx, off, 32));

    float tau_lo = mx - 1.f;
    float tau_hi = mx - exp2f(-9.f * am1);  // (1/512)^(alpha-1)

    float s = 0.f;
#pragma unroll
    for (int j = 0; j < 16; ++j) {
      float z = xs[j] - tau_lo;
      s += (z > 0.f) ? exp2f(rinv * log2f(z)) : 0.f;
    }
#pragma unroll
    for (int off = 16; off; off >>= 1) s += __shfl_xor(s, off, 32);
    const float f_lo = s - 1.f;

    float dm = tau_hi - tau_lo, tau_m = tau_lo;
    for (int it = 0; it < 50; ++it) {
      dm *= 0.5f;
      tau_m = tau_lo + dm;
      float fm = 0.f;
#pragma unroll
      for (int j = 0; j < 16; ++j) {
        float z = xs[j] - tau_m;
        fm += (z > 0.f) ? exp2f(rinv * log2f(z)) : 0.f;
      }
#pragma unroll
      for (int off = 16; off; off >>= 1) fm += __shfl_xor(fm, off, 32);
      fm -= 1.f;
      tau_lo = (fm * f_lo >= 0.f) ? tau_m : tau_lo;  // uniform update
    }

    float pv[16], ssum = 0.f;
#pragma unroll
    for (int j = 0; j < 16; ++j) {
      float z = xs[j] - tau_m;
      pv[j] = (z > 0.f) ? exp2f(rinv * log2f(z)) : 0.f;
      ssum += pv[j];
    }
#pragma unroll
    for (int off = 16; off; off >>= 1) ssum += __shfl_xor(ssum, off, 32);
    const float inv = 1.f / ssum;
#pragma unroll
    for (int j = 0; j < 16; ++j) Sc[row][lane + 32 * j] = pv[j] * inv;
  }

  // ---- phase 3: att = p @ mk ----
  v8f acc = {0.f, 0.f, 0.f, 0.f, 0.f, 0.f, 0.f, 0.f};
  for (int kb = 0; kb < LL; kb += 64) {
    __syncthreads();
#pragma unroll
    for (int i = 0; i < 8; ++i) {
      int idx = t + i * 128;
      int r = idx >> 4, c4 = (idx & 15) << 2;
      stage16(X + (size_t)(b * LL + kb + r) * DD + h * HD + c4, &Sk[r][c4]);
    }
    stage_wait();
    __syncthreads();
#pragma unroll
    for (int kk = 0; kk < 64; kk += 4) {
      v2f a, bb;
      a.x  = Sc[lm][kb + kk + lk];
      a.y  = Sc[lm][kb + kk + lk + 1];
      bb.x = Sk[kk + lk][(wave << 4) + lm];  // mk^T gather from LDS
      bb.y = Sk[kk + lk + 1][(wave << 4) + lm];
      acc = __builtin_amdgcn_wmma_f32_16x16x4_f32(false, a, false, bb,
                                                  (short)0, acc, false, false);
    }
  }
#pragma unroll
  for (int r = 0; r < 8; ++r) {
    int qr = r + ((lane >> 4) << 3);
    att[(size_t)(b * LL + q0 + qr) * DD + h * HD + (wave << 4) + lm] = acc[r];
  }
}

// ---------------------------------------------------------------------------
// LayerNorm over D=512 per row; writes split output layout:
//   rows l<511 -> out[(b*511+l)*512 + d] ; row 511 -> out[16*511*512 + b*512 + d]
// ---------------------------------------------------------------------------
__global__ __launch_bounds__(256) void ln_kernel(const float* __restrict__ Hm,
                                                 const float* __restrict__ g,
                                                 const float* __restrict__ be,
                                                 float* __restrict__ out) {
  const int l = blockIdx.x, b = blockIdx.y, t = threadIdx.x;
  const int wave = t >> 5, lane = t & 31;
  const float* hr = Hm + (size_t)(b * LL + l) * DD;
  float x0 = hr[t], x1 = hr[t + 256];

  __shared__ float red[8];
  float s = x0 + x1;
#pragma unroll
  for (int off = 16; off; off >>= 1) s += __shfl_xor(s, off, 32);
  if (lane == 0) red[wave] = s;
  __syncthreads();
  float tot = 0.f;
#pragma unroll
  for (int i = 0; i < 8; ++i) tot += red[i];
  const float mu = tot * (1.f / 512.f);

  float d0 = x0 - mu, d1 = x1 - mu;
  float vs = d0 * d0 + d1 * d1;
#pragma unroll
  for (int off = 16; off; off >>= 1) vs += __shfl_xor(vs, off, 32);
  __syncthreads();
  if (lane == 0) red[wave] = vs;
  __syncthreads();
  float vtot = 0.f;
#pragma unroll
  for (int i = 0; i < 8; ++i) vtot += red[i];
  const float inv = rsqrtf(vtot * (1.f / 512.f) + 1e-12f);

  size_t o = (l < LL - 1) ? ((size_t)b * (LL - 1) + l) * DD
                          : (size_t)BB * (LL - 1) * DD + (size_t)b * DD;
  out[o + t]       = d0 * inv * g[t] + be[t];
  out[o + t + 256] = d1 * inv * g[t + 256] + be[t + 256];
}

// ---------------------------------------------------------------------------
extern "C" void kernel_launch(void* const* d_in, const int* in_sizes, int n_in,
                              void* d_out, int out_size, void* d_ws,
                              size_t ws_size, hipStream_t stream) {
  const float* X    = (const float*)d_in[0];
  const float* mask = (const float*)d_in[1];
  const float* Wq   = (const float*)d_in[2];
  const float* bq   = (const float*)d_in[3];
  const float* W1   = (const float*)d_in[4];
  const float* b1   = (const float*)d_in[5];
  const float* W2   = (const float*)d_in[6];
  const float* b2   = (const float*)d_in[7];
  const float* ln_g = (const float*)d_in[8];
  const float* ln_b = (const float*)d_in[9];
  const float* Wa   = (const float*)d_in[10];
  const float* ba   = (const float*)d_in[11];

  float* ws = (float*)d_ws;
  const size_t NE = (size_t)BB * LL * DD;  // 4,194,304
  float* qbuf    = ws;
  float* attbuf  = ws + NE;
  float* h1buf   = ws + 2 * NE;
  float* alphaBH = ws + 3 * NE;
  float* hbuf    = qbuf;  // q dead after attention; reuse for FFN output

  const int M = BB * LL;  // 8192
  dim3 gg(M / 128, DD / 16);

  // 1. q = relu(X @ Wq^T + bq)
  gemm_wmma_kernel<true, false>
      <<<gg, 256, 0, stream>>>(X, Wq, bq, nullptr, qbuf, M, DD, DD);
  // 2. alpha per (b,h)
  alpha_kernel<<<1, 128, 0, stream>>>(qbuf, Wa, ba, alphaBH);
  // 3. scores -> entmax -> att
  attn_entmax_kernel<<<dim3(LL / 16, HEADS, BB), 128, 0, stream>>>(
      qbuf, X, mask, alphaBH, attbuf);
  // 4. h1 = relu(att @ W1^T + b1)
  gemm_wmma_kernel<true, false>
      <<<gg, 256, 0, stream>>>(attbuf, W1, b1, nullptr, h1buf, M, DD, DD);
  // 5. h = h1 @ W2^T + b2 + att
  gemm_wmma_kernel<false, true>
      <<<gg, 256, 0, stream>>>(h1buf, W2, b2, attbuf, hbuf, M, DD, DD);
  // 6. LayerNorm + split output write
  ln_kernel<<<dim3(LL, BB), 256, 0, stream>>>(hbuf, ln_g, ln_b, (float*)d_out);
}